// MHCLayer_82970178224517
// MI455X (gfx1250) — compile-verified
//
#include <hip/hip_runtime.h>

#define EPSV 1e-5f

typedef float v2f __attribute__((ext_vector_type(2)));
typedef float v4f __attribute__((ext_vector_type(4)));
typedef float v8f __attribute__((ext_vector_type(8)));
typedef int   v4i __attribute__((ext_vector_type(4)));

// ---------------------------------------------------------------------------
// CDNA5 async global->LDS copy (ASYNCcnt) with sync fallback
// ---------------------------------------------------------------------------
#if __has_builtin(__builtin_amdgcn_global_load_async_to_lds_b128)
#define USE_ASYNC 1
typedef __attribute__((address_space(1))) v4i as1_v4i;
typedef __attribute__((address_space(3))) v4i as3_v4i;
__device__ __forceinline__ void async16(const v4f* g, float* l) {
    __builtin_amdgcn_global_load_async_to_lds_b128((as1_v4i*)g, (as3_v4i*)l, 0, 0);
}
#if __has_builtin(__builtin_amdgcn_s_wait_asynccnt)
#define WAIT_ASYNC(n) __builtin_amdgcn_s_wait_asynccnt(n)
#else
#define WAIT_ASYNC(n) asm volatile("s_wait_asynccnt %0" ::"n"(n) : "memory")
#endif
#else
#define USE_ASYNC 0
__device__ __forceinline__ void async16(const v4f* g, float* l) { *(v4f*)l = *g; }
#define WAIT_ASYNC(n)
#endif

// ---------------------------------------------------------------------------
// Kernel 0: phiw[32][8192] = phi * rmsnorm_weight (rows 24..31 zero-padded)
// ---------------------------------------------------------------------------
__global__ __launch_bounds__(256)
void mhc_phiw_kernel(const float* __restrict__ phi_pre,
                     const float* __restrict__ phi_post,
                     const float* __restrict__ phi_res,
                     const float* __restrict__ wgt,
                     float* __restrict__ phiw)
{
    int idx = blockIdx.x * 256 + threadIdx.x;   // one float4 each, 65536 total
    int row = idx >> 11;                        // 2048 float4 per row
    int col = idx & 2047;
    v4f v = {0.f, 0.f, 0.f, 0.f};
    if (row < 24) {
        const float* prow = (row < 4)  ? phi_pre  + (size_t)row * 8192
                          : (row < 8)  ? phi_post + (size_t)(row - 4) * 8192
                                       : phi_res  + (size_t)(row - 8) * 8192;
        v = ((const v4f*)prow)[col] * ((const v4f*)wgt)[col & 511];
    }
    ((v4f*)phiw)[idx] = v;
}

// ---------------------------------------------------------------------------
// Kernel 1: fused RMS-stats + projection GEMM  proj[B,24] = rmsnorm(x) @ phi^T
// A = phiw rows (M, 2 tiles of 16), B = x (N = 16 batches), K = channels.
// Double-buffered async global->LDS staging; inv_rms applied per-stream.
// ---------------------------------------------------------------------------
#define KC 256
#define PITCH 260       // +4 floats -> conflict-free 16-row fragment reads

__global__ __launch_bounds__(256)
void mhc_proj_kernel(const float* __restrict__ x,
                     const float* __restrict__ phiw,
                     float* __restrict__ proj)
{
    __shared__ float xs[2 * 16 * PITCH];    // double-buffered x chunk
    __shared__ float ph[2 * 32 * PITCH];    // double-buffered phiw chunk
    __shared__ float S_lds[16];
    __shared__ float inv_lds[16];
    __shared__ float proj_lds[32 * 16];

    const int t     = threadIdx.x;
    const int tile  = blockIdx.x;           // 16 batches per block
    const int lane  = t & 31;
    const int wave  = t >> 5;
    const int half  = lane >> 4;
    const int l     = lane & 15;
    const int mtile = wave >> 2;            // phi-row tile
    const int ksub  = wave & 3;             // 64-wide K sub-range in chunk

    if (t < 16) S_lds[t] = 0.0f;
    for (int i = t; i < 512; i += 256) proj_lds[i] = 0.0f;

    // ---- staging issue: 4 x-float4 + 8 phi-float4 per thread (12 async/wave)
    auto stage_issue = [&](int gc, int buf) {
        const int kbase = gc * KC;          // gc spans all 4 streams (0..31)
        float* xsb = xs + buf * (16 * PITCH);
        float* phb = ph + buf * (32 * PITCH);
#pragma unroll
        for (int i = 0; i < 4; ++i) {
            int f4 = i * 256 + t, row = f4 >> 6, col = f4 & 63;
            const v4f* g = (const v4f*)(x + (size_t)(tile * 16 + row) * 8192 + kbase) + col;
            async16(g, &xsb[row * PITCH + col * 4]);
        }
#pragma unroll
        for (int i = 0; i < 8; ++i) {
            int f4 = i * 256 + t, row = f4 >> 6, col = f4 & 63;
            const v4f* g = (const v4f*)(phiw + (size_t)row * 8192 + kbase) + col;
            async16(g, &phb[row * PITCH + col * 4]);
        }
    };

    v8f projacc = {0.f, 0.f, 0.f, 0.f, 0.f, 0.f, 0.f, 0.f};
    v8f P       = {0.f, 0.f, 0.f, 0.f, 0.f, 0.f, 0.f, 0.f};
    float ssq = 0.0f;

    stage_issue(0, 0);                      // prologue

    for (int gc = 0; gc < 32; ++gc) {       // 32 chunks = 4 streams * 8
        const int cur = gc & 1;
        if (gc < 31) { stage_issue(gc + 1, cur ^ 1); WAIT_ASYNC(12); }
        else         { WAIT_ASYNC(0); }
        __syncthreads();                    // chunk gc resident in buffer cur

        const float* xsb = xs + cur * (16 * PITCH);
        const float* phb = ph + cur * (32 * PITCH);
        const int kb = ksub * 64;
#pragma unroll
        for (int s = 0; s < 16; ++s) {
            int k = kb + s * 4 + 2 * half;
            v2f a = *(const v2f*)&phb[(mtile * 16 + l) * PITCH + k]; // phi rows
            v2f b = *(const v2f*)&xsb[l * PITCH + k];                // x batches
            ssq = fmaf(b.x, b.x, fmaf(b.y, b.y, ssq));               // free harvest
            P = __builtin_amdgcn_wmma_f32_16x16x4_f32(
                    false, a, false, b, (short)0, P, false, false);
        }
        __syncthreads();                    // buffer reused next iteration

        if ((gc & 7) == 7) {                // ---- stream epilogue ----
            if (mtile == 0) atomicAdd(&S_lds[l], ssq);  // waves 0-3 cover full K
            __syncthreads();
            if (t < 16) {
                inv_lds[t] = __frsqrt_rn(S_lds[t] * (1.0f / 2048.0f) + EPSV);
                S_lds[t] = 0.0f;
            }
            __syncthreads();
            float f = inv_lds[l];           // C cols = batches -> per-lane scale
#pragma unroll
            for (int r = 0; r < 8; ++r) projacc[r] = fmaf(P[r], f, projacc[r]);
            P = (v8f){0.f, 0.f, 0.f, 0.f, 0.f, 0.f, 0.f, 0.f};
            ssq = 0.0f;
        }
    }
    // ---- reduce the 4 K-sub-range waves per mtile ----
#pragma unroll
    for (int r = 0; r < 8; ++r) {
        int row = mtile * 16 + r + 8 * half;
        atomicAdd(&proj_lds[row * 16 + l], projacc[r]);
    }
    __syncthreads();
    if (t < 384) {                          // 24 rows x 16 batches
        int r = t >> 4, b = t & 15;
        proj[(size_t)(tile * 16 + b) * 24 + r] = proj_lds[r * 16 + b];
    }
}

// ---------------------------------------------------------------------------
// Kernel 2: per-batch Sinkhorn (4x4, 20 iters, lane-shuffle) + 4x4 @ 4x2048 mix
// ---------------------------------------------------------------------------
__global__ __launch_bounds__(256)
void mhc_mix_kernel(const float* __restrict__ x,
                    const float* __restrict__ proj,
                    const float* __restrict__ b_pre,
                    const float* __restrict__ b_post,
                    const float* __restrict__ b_res,
                    const float* __restrict__ a_pre,
                    const float* __restrict__ a_post,
                    const float* __restrict__ a_res,
                    float* __restrict__ out)
{
    __shared__ float Wm[16];
    const int t = threadIdx.x;
    const size_t b = blockIdx.x;

    if (t < 32) {                                   // wave 0 does Sinkhorn
        const int li = t & 15;
        const int i = li >> 2, j = li & 3;
        float logit = fmaf(a_res[0], proj[b * 24 + 8 + li], b_res[li]);
        float m = logit;
        m = fmaxf(m, __shfl_xor(m, 1, 32));
        m = fmaxf(m, __shfl_xor(m, 2, 32));
        m = fmaxf(m, __shfl_xor(m, 4, 32));
        m = fmaxf(m, __shfl_xor(m, 8, 32));
        float v = __expf(logit - m);
#pragma unroll
        for (int it = 0; it < 20; ++it) {
            float rs = v + __shfl_xor(v, 1, 32);    // row sum (over j)
            rs += __shfl_xor(rs, 2, 32);
            v = v / (rs + EPSV);
            float cs = v + __shfl_xor(v, 4, 32);    // col sum (over i)
            cs += __shfl_xor(cs, 8, 32);
            v = v / (cs + EPSV);
        }
        float hpre  = fmaf(a_pre[0],  proj[b * 24 + j],     b_pre[j]);
        float hpost = fmaf(a_post[0], proj[b * 24 + 4 + i], b_post[i]);
        if (t < 16) Wm[li] = fmaf(hpost, hpre, v);
    }
    __syncthreads();

    float W[16];
#pragma unroll
    for (int r = 0; r < 16; ++r) W[r] = Wm[r];

    const v4f* xb = (const v4f*)(x + b * 8192);
    v4f* ob = (v4f*)(out + b * 8192);
#pragma unroll
    for (int cc = 0; cc < 2; ++cc) {
        int c4 = t + cc * 256;
        v4f x0 = xb[c4];
        v4f x1 = xb[512 + c4];
        v4f x2 = xb[1024 + c4];
        v4f x3 = xb[1536 + c4];
#pragma unroll
        for (int i = 0; i < 4; ++i) {
            v4f o = W[i * 4 + 0] * x0 + W[i * 4 + 1] * x1
                  + W[i * 4 + 2] * x2 + W[i * 4 + 3] * x3;
            __builtin_nontemporal_store(o, &ob[i * 512 + c4]);
        }
    }
}

// ---------------------------------------------------------------------------
extern "C" void kernel_launch(void* const* d_in, const int* in_sizes, int n_in,
                              void* d_out, int out_size, void* d_ws, size_t ws_size,
                              hipStream_t stream) {
    const float* x        = (const float*)d_in[0];
    const float* wgt      = (const float*)d_in[1];
    const float* phi_pre  = (const float*)d_in[2];
    const float* phi_post = (const float*)d_in[3];
    const float* phi_res  = (const float*)d_in[4];
    const float* b_pre    = (const float*)d_in[5];
    const float* b_post   = (const float*)d_in[6];
    const float* b_res    = (const float*)d_in[7];
    const float* a_pre    = (const float*)d_in[8];
    const float* a_post   = (const float*)d_in[9];
    const float* a_res    = (const float*)d_in[10];
    float* out  = (float*)d_out;

    float* phiw = (float*)d_ws;                 // 32 x 8192 floats (1 MB)
    float* proj = phiw + 32 * 8192;             // B x 24 floats (768 KB)

    const int B = in_sizes[0] / (4 * 2048);

    hipLaunchKernelGGL(mhc_phiw_kernel, dim3(256), dim3(256), 0, stream,
                       phi_pre, phi_post, phi_res, wgt, phiw);
    hipLaunchKernelGGL(mhc_proj_kernel, dim3(B / 16), dim3(256), 0, stream,
                       x, phiw, proj);
    hipLaunchKernelGGL(mhc_mix_kernel, dim3(B), dim3(256), 0, stream,
                       x, proj, b_pre, b_post, b_res,
                       a_pre, a_post, a_res, out);
}